// SegmentLinearEmbedMovingWindow_66649302499514
// MI455X (gfx1250) — compile-verified
//
#include <hip/hip_runtime.h>
#include <hip/hip_bf16.h>

// Problem constants
#define BB   64
#define TT   4320
#define CC   64
#define SEG  24
#define NSEG 180            // TT / SEG
#define EE   512
#define MM   (BB * NSEG)    // 11520
#define KK   (CC * SEG)     // 1536
#define EPS  1e-6f

// GEMM tiling
#define TM   128
#define TN   64
#define TK   16
#define NST  (KK / TK)      // 96 k-stages
#define LDP  20             // padded LDS row stride (floats): 16B-aligned, bank-conflict-free

typedef __attribute__((ext_vector_type(2))) float v2f;
typedef __attribute__((ext_vector_type(8))) float v8f;

// ---------------------------------------------------------------------------
// CDNA5 async global->LDS helpers (ASYNCcnt-tracked, no VGPR transit)
// ---------------------------------------------------------------------------
__device__ __forceinline__ unsigned lds_addr32(const void* p) {
    // Flat LDS aperture: hardware uses addr[31:0] as the LDS byte address.
    return (unsigned)(reinterpret_cast<uintptr_t>(p));
}

__device__ __forceinline__ void async_copy_b128(unsigned lds, const float* g) {
    asm volatile("global_load_async_to_lds_b128 %0, %1, off"
                 :: "v"(lds), "v"(g)
                 : "memory");
}

template <int N>
__device__ __forceinline__ void wait_async_le() {
#if __has_builtin(__builtin_amdgcn_s_wait_asynccnt)
    __builtin_amdgcn_s_wait_asynccnt(N);
#else
    asm volatile("s_wait_asynccnt %0" :: "n"(N) : "memory");
#endif
}

// ---------------------------------------------------------------------------
// Kernel 1: per-(segment, feature) normalization, packed into A[M][K], k=c*24+s
// ---------------------------------------------------------------------------
__global__ void norm_pack_kernel(const float* __restrict__ x,
                                 float* __restrict__ Xn) {
    int gid = blockIdx.x * blockDim.x + threadIdx.x;   // over MM*CC
    int c = gid & (CC - 1);
    int m = gid >> 6;                                  // bn index
    if (m >= MM) return;

    const long xbase = (long)m * SEG * CC + c;         // x[(m*24 + s)*64 + c]
    float v[SEG];
    float sum = 0.0f;
#pragma unroll
    for (int s = 0; s < SEG; ++s) {
        v[s] = x[xbase + (long)s * CC];
        sum += v[s];
    }
    float mean = sum * (1.0f / SEG);
    float var = 0.0f;
#pragma unroll
    for (int s = 0; s < SEG; ++s) {
        float d = v[s] - mean;
        var += d * d;
    }
    var *= (1.0f / (SEG - 1));                         // unbiased (ddof=1)
    float inv = 1.0f / (__builtin_sqrtf(var) + EPS);

    float* dst = Xn + (long)m * KK + c * SEG;          // contiguous 24 floats
#pragma unroll
    for (int s = 0; s < SEG; ++s) {
        dst[s] = (v[s] - mean) * inv;
    }
}

// ---------------------------------------------------------------------------
// Kernel 2: repack W[c,e,s] -> Wp[e][k], k = c*24 + s  (N-major, K-contiguous)
// ---------------------------------------------------------------------------
__global__ void wpack_kernel(const float* __restrict__ W,
                             float* __restrict__ Wp) {
    int gid = blockIdx.x * blockDim.x + threadIdx.x;   // over EE*CC
    int c = gid & (CC - 1);
    int e = gid >> 6;
    if (e >= EE) return;
    const float* src = W + ((long)c * EE + e) * SEG;   // contiguous in s
    float*       dst = Wp + (long)e * KK + c * SEG;
#pragma unroll
    for (int s = 0; s < SEG; ++s) dst[s] = src[s];
}

// ---------------------------------------------------------------------------
// Kernel 3: bm[e] = mean_c b[c,e]
// ---------------------------------------------------------------------------
__global__ void bmean_kernel(const float* __restrict__ b,
                             float* __restrict__ bm) {
    int e = blockIdx.x * blockDim.x + threadIdx.x;
    if (e >= EE) return;
    float s = 0.0f;
#pragma unroll 8
    for (int c = 0; c < CC; ++c) s += b[(long)c * EE + e];
    bm[e] = s * (1.0f / CC);
}

// ---------------------------------------------------------------------------
// Kernel 4: WMMA f32 GEMM:  out[m][e] = (1/C) * Xn[m][:] . Wp[e][:]  + bm[e]
// 128x64 block tile, 8 wave32 waves, 32x32 per wave (2x2 wmma accumulators).
// Double-buffered LDS fed by GLOBAL_LOAD_ASYNC_TO_LDS_B128 (ASYNCcnt).
// ---------------------------------------------------------------------------
__global__ void __launch_bounds__(256)
wmma_gemm_kernel(const float* __restrict__ A,     // Xn [MM][KK]
                 const float* __restrict__ Bp,    // Wp [EE][KK]
                 const float* __restrict__ bm,    // [EE]
                 float* __restrict__ out) {       // [MM][EE]
    __shared__ float As[2][TM][LDP];              // [buf][m][k]
    __shared__ float Bs[2][TN][LDP];              // [buf][n][k] (transposed tile)

    const int tid  = threadIdx.x;
    const int wave = tid >> 5;
    const int lane = tid & 31;
    const int half = lane >> 4;
    const int ln   = lane & 15;

    const int m0 = blockIdx.x * TM;               // 90 tiles
    const int n0 = blockIdx.y * TN;               // 8 tiles
    const int waveM = (wave >> 1) * 32;           // 4 waves along M
    const int waveN = (wave & 1) * 32;            // 2 waves along N

    // Per-thread staging coordinates: 16B chunks.
    const int rowA = tid >> 2;                    // 0..63   (second A chunk: +64)
    const int colA = (tid & 3) << 2;              // 0,4,8,12
    const float* gA0 = A  + (long)(m0 + rowA)      * KK + colA;
    const float* gA1 = A  + (long)(m0 + rowA + 64) * KK + colA;
    const float* gB  = Bp + (long)(n0 + rowA)      * KK + colA;

    const unsigned lA0[2] = { lds_addr32(&As[0][rowA][colA]),
                              lds_addr32(&As[1][rowA][colA]) };
    const unsigned lA1[2] = { lds_addr32(&As[0][rowA + 64][colA]),
                              lds_addr32(&As[1][rowA + 64][colA]) };
    const unsigned lB [2] = { lds_addr32(&Bs[0][rowA][colA]),
                              lds_addr32(&Bs[1][rowA][colA]) };

    v8f acc00 = {}, acc01 = {}, acc10 = {}, acc11 = {};

    // Prologue: stage 0 into buffer 0 (3 async B128 per thread).
    async_copy_b128(lA0[0], gA0);
    async_copy_b128(lA1[0], gA1);
    async_copy_b128(lB[0],  gB);

    for (int s = 0; s < NST; ++s) {
        const int cur = s & 1;
        if (s + 1 < NST) {
            const int nxt = cur ^ 1;
            const int k1 = (s + 1) * TK;
            async_copy_b128(lA0[nxt], gA0 + k1);
            async_copy_b128(lA1[nxt], gA1 + k1);
            async_copy_b128(lB[nxt],  gB + k1);
            wait_async_le<3>();                   // stage s's 3 copies complete
        } else {
            wait_async_le<0>();
        }
        __syncthreads();                          // all waves' LDS writes visible

        // ---- 4 k-steps of V_WMMA_F32_16X16X4_F32 per wave sub-tile ----
#pragma unroll
        for (int kk = 0; kk < TK; kk += 4) {
            const int kc = kk + (half << 1);      // k = kk + 2*half (+0/+1)
            v2f a0 = *(const v2f*)&As[cur][waveM +      ln][kc];
            v2f a1 = *(const v2f*)&As[cur][waveM + 16 + ln][kc];
            v2f b0 = *(const v2f*)&Bs[cur][waveN +      ln][kc];
            v2f b1 = *(const v2f*)&Bs[cur][waveN + 16 + ln][kc];
            acc00 = __builtin_amdgcn_wmma_f32_16x16x4_f32(false, a0, false, b0,
                                                          (short)0, acc00, false, false);
            acc01 = __builtin_amdgcn_wmma_f32_16x16x4_f32(false, a0, false, b1,
                                                          (short)0, acc01, false, false);
            acc10 = __builtin_amdgcn_wmma_f32_16x16x4_f32(false, a1, false, b0,
                                                          (short)0, acc10, false, false);
            acc11 = __builtin_amdgcn_wmma_f32_16x16x4_f32(false, a1, false, b1,
                                                          (short)0, acc11, false, false);
        }
        __syncthreads();                          // safe to overwrite buf at s+2
    }

    // ---- epilogue: D layout (ISA 7.12.2): m = r + 8*half, n = ln ----
    const float scale = 1.0f / CC;
    const int e0 = n0 + waveN + ln;               // j = 0
    const int e1 = e0 + 16;                       // j = 1
    const float bm0 = bm[e0];
    const float bm1 = bm[e1];
    const int mb = m0 + waveM + (half << 3);
#pragma unroll
    for (int r = 0; r < 8; ++r) {
        const long row0 = (long)(mb + r) * EE;        // i = 0
        const long row1 = (long)(mb + 16 + r) * EE;   // i = 1
        out[row0 + e0] = acc00[r] * scale + bm0;
        out[row0 + e1] = acc01[r] * scale + bm1;
        out[row1 + e0] = acc10[r] * scale + bm0;
        out[row1 + e1] = acc11[r] * scale + bm1;
    }
}

// ---------------------------------------------------------------------------
extern "C" void kernel_launch(void* const* d_in, const int* in_sizes, int n_in,
                              void* d_out, int out_size, void* d_ws, size_t ws_size,
                              hipStream_t stream) {
    const float* x = (const float*)d_in[0];   // [64,4320,64]
    const float* W = (const float*)d_in[1];   // [64,512,24]
    const float* b = (const float*)d_in[2];   // [64,512]
    float* out = (float*)d_out;               // [64,180,512]

    char* ws = (char*)d_ws;
    const size_t XN_BYTES = (size_t)MM * KK * sizeof(float);   // 70.8 MB
    const size_t WP_BYTES = (size_t)EE * KK * sizeof(float);   // 3.1 MB
    float* Xn = (float*)(ws);
    float* Wp = (float*)(ws + XN_BYTES);
    float* bm = (float*)(ws + XN_BYTES + WP_BYTES);

    // 1) normalize + pack A
    norm_pack_kernel<<<(MM * CC) / 256, 256, 0, stream>>>(x, Xn);
    // 2) repack W
    wpack_kernel<<<(EE * CC) / 256, 256, 0, stream>>>(W, Wp);
    // 3) bias mean
    bmean_kernel<<<EE / 256, 256, 0, stream>>>(b, bm);
    // 4) WMMA GEMM + epilogue (async double-buffered)
    dim3 grid(MM / TM, EE / TN);   // (90, 8)
    wmma_gemm_kernel<<<grid, 256, 0, stream>>>(Xn, Wp, bm, out);
}